// GraphDenoisingModel_3178275799464
// MI455X (gfx1250) — compile-verified
//
#include <hip/hip_runtime.h>
#include <hip/hip_bf16.h>
#include <math.h>

// ---------------------------------------------------------------------------
// Problem constants (match reference)
// ---------------------------------------------------------------------------
#define NB   8192   // batch N
#define CIN  32
#define LLEN 64
#define DD   64

typedef __attribute__((ext_vector_type(2))) float v2f;
typedef __attribute__((ext_vector_type(8))) float v8f;
typedef int i32x4 __attribute__((vector_size(16)));   // matches builtin param type

__device__ inline v8f v8f_zero() {
    v8f z = {0.f, 0.f, 0.f, 0.f, 0.f, 0.f, 0.f, 0.f};
    return z;
}

// D = A(16x4 f32) x B(4x16 f32) + C(16x16 f32), wave32, all lanes active.
__device__ inline v8f wmma_f32_16x16x4(v2f a, v2f b, v8f c) {
#if __has_builtin(__builtin_amdgcn_wmma_f32_16x16x4_f32)
    return __builtin_amdgcn_wmma_f32_16x16x4_f32(
        /*neg_a=*/false, a, /*neg_b=*/false, b,
        /*c_mod=*/(short)0, c, /*reuse_a=*/false, /*reuse_b=*/false);
#else
    c[0] += a.x * b.x;   // compile-safe marker path (wmma=0 signals this)
    return c;
#endif
}

// Async global -> LDS 16-byte copy (GLOBAL_LOAD_ASYNC_TO_LDS_B128, ASYNCcnt).
// Builtin parameter types per hipcc diagnostic: (AS1 i32x4*, AS3 i32x4*, Ii, Ii).
__device__ inline void async_copy_f4(const float* gsrc, float* lds_dst) {
#if __has_builtin(__builtin_amdgcn_global_load_async_to_lds_b128)
    __builtin_amdgcn_global_load_async_to_lds_b128(
        (__attribute__((address_space(1))) i32x4*)(i32x4*)(float*)gsrc,
        (__attribute__((address_space(3))) i32x4*)(i32x4*)lds_dst,
        /*imm offset=*/0, /*cpol=*/0);
#else
    *(float4*)lds_dst = *(const float4*)gsrc;
#endif
}

__device__ inline void wait_async_zero() {
#if __has_builtin(__builtin_amdgcn_s_wait_asynccnt)
    __builtin_amdgcn_s_wait_asynccnt(0);
#else
    asm volatile("s_wait_asynccnt 0" ::: "memory");
#endif
}

// ---------------------------------------------------------------------------
// Kernel 1: per-sample feature extractor.
// conv1/conv2 as implicit GEMM via V_WMMA_F32_16X16X4_F32, fused relu,
// mean pool, fc1, fc2.  One 128-thread (4-wave) block per sample.
// ---------------------------------------------------------------------------
__global__ __launch_bounds__(128)
void feat_kernel(const float* __restrict__ x,     // [N][32][64]
                 const float* __restrict__ w1,    // [64][32][3]  (row r = c*3+k)
                 const float* __restrict__ b1,    // [64]
                 const float* __restrict__ w2,    // [128][64][3]
                 const float* __restrict__ b2,    // [128]
                 const float* __restrict__ fc1w,  // [256][128]
                 const float* __restrict__ fc1b,  // [256]
                 const float* __restrict__ fc2w,  // [64][256]
                 const float* __restrict__ fc2b,  // [64]
                 float* __restrict__ Fo,          // ws   [N][64]
                 float* __restrict__ fout)        // dout [N][64] (tail of d_out)
{
    const int n    = blockIdx.x;
    const int tid  = threadIdx.x;
    const int wave = tid >> 5;
    const int lane = tid & 31;
    const int lm   = lane & 15;   // M (or N) index within half-wave
    const int lh   = lane >> 4;   // half select (K pair select)

    __shared__ float xp[CIN][LLEN + 2];   // zero-padded input   (8.4 KB)
    __shared__ float h1[64][LLEN + 2];    // zero-padded conv1   (16.9 KB)
    __shared__ float pooled[128];
    __shared__ float hfc[256];

    // Load x with 'same' padding (border columns = 0).
    for (int idx = tid; idx < CIN * (LLEN + 2); idx += 128) {
        const int c = idx / (LLEN + 2);
        const int l = idx % (LLEN + 2);
        float v = 0.f;
        if (l >= 1 && l <= LLEN)
            v = x[(size_t)n * (CIN * LLEN) + c * LLEN + (l - 1)];
        xp[c][l] = v;
    }
    for (int o = tid; o < 64; o += 128) { h1[o][0] = 0.f; h1[o][LLEN + 1] = 0.f; }
    __syncthreads();

    // ---- conv1: out[64][64] = W1[64][96] x B[96][64],  B[r][l] = xp[r/3][l + r%3]
    {
        const int m0 = wave * 16;             // each wave: one M-tile, all 4 N-tiles
        v8f acc[4];
#pragma unroll
        for (int t = 0; t < 4; ++t) acc[t] = v8f_zero();

        for (int kk = 0; kk < 24; ++kk) {
            const int ka = kk * 4 + 2 * lh;   // this lane's first K
            v2f a;
            a.x = w1[(m0 + lm) * 96 + ka];
            a.y = w1[(m0 + lm) * 96 + ka + 1];
#pragma unroll
            for (int t = 0; t < 4; ++t) {
                const int n0 = t * 16;
                v2f b;
                { int r = ka;     b.x = xp[r / 3][n0 + lm + (r % 3)]; }
                { int r = ka + 1; b.y = xp[r / 3][n0 + lm + (r % 3)]; }
                acc[t] = wmma_f32_16x16x4(a, b, acc[t]);
            }
        }
        // epilogue: bias + relu -> padded h1
#pragma unroll
        for (int t = 0; t < 4; ++t) {
#pragma unroll
            for (int v = 0; v < 8; ++v) {
                const int o = m0 + v + 8 * lh;
                const int l = t * 16 + lm;
                h1[o][l + 1] = fmaxf(acc[t][v] + b1[o], 0.f);
            }
        }
    }
    __syncthreads();

    // ---- conv2 + mean pool: out[128][64] = W2[128][192] x B[192][64]
    for (int mt = 0; mt < 2; ++mt) {
        const int m0 = (wave * 2 + mt) * 16;
        v8f acc[4];
#pragma unroll
        for (int t = 0; t < 4; ++t) acc[t] = v8f_zero();

        for (int kk = 0; kk < 48; ++kk) {
            const int ka = kk * 4 + 2 * lh;
            v2f a;
            a.x = w2[(m0 + lm) * 192 + ka];
            a.y = w2[(m0 + lm) * 192 + ka + 1];
#pragma unroll
            for (int t = 0; t < 4; ++t) {
                const int n0 = t * 16;
                v2f b;
                { int r = ka;     b.x = h1[r / 3][n0 + lm + (r % 3)]; }
                { int r = ka + 1; b.y = h1[r / 3][n0 + lm + (r % 3)]; }
                acc[t] = wmma_f32_16x16x4(a, b, acc[t]);
            }
        }
        // epilogue: bias, relu, mean over l; lanes 0..15 of each half share one row o
#pragma unroll
        for (int v = 0; v < 8; ++v) {
            const int o = m0 + v + 8 * lh;
            float s = 0.f;
#pragma unroll
            for (int t = 0; t < 4; ++t) s += fmaxf(acc[t][v] + b2[o], 0.f);
            for (int off = 8; off >= 1; off >>= 1) s += __shfl_xor(s, off, 16);
            if (lm == 0) pooled[o] = s * (1.f / 64.f);
        }
    }
    __syncthreads();

    // ---- fc1 (relu)
    for (int o = tid; o < 256; o += 128) {
        float s = fc1b[o];
        for (int e = 0; e < 128; ++e) s += fc1w[o * 128 + e] * pooled[e];
        hfc[o] = fmaxf(s, 0.f);
    }
    __syncthreads();

    // ---- fc2 -> f[n]
    if (tid < 64) {
        float s = fc2b[tid];
        for (int e = 0; e < 256; ++e) s += fc2w[tid * 256 + e] * hfc[e];
        Fo  [(size_t)n * DD + tid] = s;
        fout[(size_t)n * DD + tid] = s;
    }
}

// ---------------------------------------------------------------------------
// Kernel 2: M = Q Q^T  (64x64, trivial)
// ---------------------------------------------------------------------------
__global__ void mmat_kernel(const float* __restrict__ Q, float* __restrict__ M) {
    const int tid = threadIdx.x;   // single 128-thread block
    for (int idx = tid; idx < 64 * 64; idx += 128) {
        const int i = idx >> 6, j = idx & 63;
        float s = 0.f;
        for (int k = 0; k < 64; ++k) s += Q[i * 64 + k] * Q[j * 64 + k];
        M[idx] = s;
    }
}

// ---------------------------------------------------------------------------
// Kernel 3: P = F M, diag[n] = f_n . p_n, zero rowsum.
// ---------------------------------------------------------------------------
__global__ __launch_bounds__(128)
void proj_kernel(const float* __restrict__ F, const float* __restrict__ M,
                 float* __restrict__ P, float* __restrict__ diag,
                 float* __restrict__ rowsum) {
    __shared__ float Ms[64][64];
    const int tid = threadIdx.x;
    for (int idx = tid; idx < 64 * 64; idx += 128)
        Ms[idx >> 6][idx & 63] = M[idx];
    __syncthreads();

    const int n = blockIdx.x * 128 + tid;   // grid = N/128 blocks
    float fr[64];
#pragma unroll
    for (int e = 0; e < 64; ++e) fr[e] = F[(size_t)n * 64 + e];
    float dg = 0.f;
    for (int d = 0; d < 64; ++d) {
        float s = 0.f;
#pragma unroll
        for (int e = 0; e < 64; ++e) s += fr[e] * Ms[e][d];
        P[(size_t)n * 64 + d] = s;
        dg += fr[d] * s;
    }
    diag[n]   = dg;
    rowsum[n] = 0.f;
}

// ---------------------------------------------------------------------------
// Kernel 4: pairwise Gram tile + Laplacian epilogue.
// Block = 128 threads (4 waves) -> 64x64 tile of Lb.  F/P tiles staged via
// GLOBAL_LOAD_ASYNC_TO_LDS_B128; G = F . P^T via WMMA; then
// d_ij = diag_i + diag_j - 2g, w = b_i b_j exp(-d); Lb = -w (non-temporal,
// 256 MB stream doesn't fit L2), rowsum += w.
// ---------------------------------------------------------------------------
#define TP 68   // LDS row pitch (floats): 272 B = 16B-aligned, bank stride 4

__global__ __launch_bounds__(128)
void pair_kernel(const float* __restrict__ F, const float* __restrict__ P,
                 const float* __restrict__ diag, const float* __restrict__ beta,
                 float* __restrict__ Lb, float* __restrict__ rowsum) {
    const int i0   = blockIdx.y * 64;
    const int j0   = blockIdx.x * 64;
    const int tid  = threadIdx.x;
    const int wave = tid >> 5;
    const int lane = tid & 31;
    const int lm   = lane & 15;
    const int lh   = lane >> 4;

    __shared__ __align__(16) float Fs[64][TP];
    __shared__ __align__(16) float Ps[64][TP];

    // Stage both 64x64 tiles with async global->LDS B128 copies.
    for (int idx = tid; idx < 64 * 16; idx += 128) {
        const int r  = idx >> 4;          // row 0..63
        const int c4 = (idx & 15) * 4;    // 4-float chunk
        async_copy_f4(&F[(size_t)(i0 + r) * 64 + c4], &Fs[r][c4]);
        async_copy_f4(&P[(size_t)(j0 + r) * 64 + c4], &Ps[r][c4]);
    }
    wait_async_zero();
    __syncthreads();

    const int iw = wave * 16;      // wave's 16-row strip
    v8f acc[4];
#pragma unroll
    for (int t = 0; t < 4; ++t) acc[t] = v8f_zero();

    for (int kk = 0; kk < 16; ++kk) {
        const int ka = kk * 4 + 2 * lh;
        v2f a;                                    // A = F strip, 16x4 chunk
        a.x = Fs[iw + lm][ka];
        a.y = Fs[iw + lm][ka + 1];
#pragma unroll
        for (int t = 0; t < 4; ++t) {             // B = P^T chunk: B[k][n] = P[j0+n][k]
            v2f b;
            b.x = Ps[t * 16 + lm][ka];
            b.y = Ps[t * 16 + lm][ka + 1];
            acc[t] = wmma_f32_16x16x4(a, b, acc[t]);
        }
    }

    // epilogue
    float dj[4], bj[4];
#pragma unroll
    for (int t = 0; t < 4; ++t) {
        const int j = j0 + t * 16 + lm;
        dj[t] = diag[j];
        bj[t] = beta[j];
    }
#pragma unroll
    for (int v = 0; v < 8; ++v) {
        const int   i   = i0 + iw + v + 8 * lh;
        const float di  = diag[i];
        const float bti = beta[i];
        float rs = 0.f;
#pragma unroll
        for (int t = 0; t < 4; ++t) {
            const int   j = j0 + t * 16 + lm;
            const float g = acc[t][v];
            const float d = di + dj[t] - 2.f * g;
            const float w = bti * bj[t] * __expf(-d);
            __builtin_nontemporal_store(-w, &Lb[(size_t)i * NB + j]);
            rs += w;
        }
        for (int off = 8; off >= 1; off >>= 1) rs += __shfl_xor(rs, off, 16);
        if (lm == 0) atomicAdd(&rowsum[i], rs);
    }
}

// ---------------------------------------------------------------------------
// Kernel 5: Lb[i][i] += rowsum[i]
// ---------------------------------------------------------------------------
__global__ void diagfix_kernel(const float* __restrict__ rowsum,
                               float* __restrict__ Lb) {
    const int i = blockIdx.x * 256 + threadIdx.x;
    if (i < NB) Lb[(size_t)i * NB + i] += rowsum[i];
}

// ---------------------------------------------------------------------------
extern "C" void kernel_launch(void* const* d_in, const int* in_sizes, int n_in,
                              void* d_out, int out_size, void* d_ws, size_t ws_size,
                              hipStream_t stream) {
    const float* x    = (const float*)d_in[0];
    const float* beta = (const float*)d_in[1];
    const float* w1   = (const float*)d_in[2];
    const float* b1   = (const float*)d_in[3];
    const float* w2   = (const float*)d_in[4];
    const float* b2   = (const float*)d_in[5];
    const float* fc1w = (const float*)d_in[6];
    const float* fc1b = (const float*)d_in[7];
    const float* fc2w = (const float*)d_in[8];
    const float* fc2b = (const float*)d_in[9];
    const float* Q    = (const float*)d_in[10];

    float* out  = (float*)d_out;
    float* Lb   = out;                         // [N][N]
    float* fout = out + (size_t)NB * NB;       // [N][64]

    float* ws     = (float*)d_ws;
    float* F      = ws;                        // N*64
    float* P      = F + (size_t)NB * 64;       // N*64
    float* M      = P + (size_t)NB * 64;       // 64*64
    float* diag   = M + 64 * 64;               // N
    float* rowsum = diag + NB;                 // N

    feat_kernel<<<NB, 128, 0, stream>>>(x, w1, b1, w2, b2,
                                        fc1w, fc1b, fc2w, fc2b, F, fout);
    mmat_kernel<<<1, 128, 0, stream>>>(Q, M);
    proj_kernel<<<NB / 128, 128, 0, stream>>>(F, M, P, diag, rowsum);

    dim3 grid(NB / 64, NB / 64);
    pair_kernel<<<grid, 128, 0, stream>>>(F, P, diag, beta, Lb, rowsum);
    diagfix_kernel<<<(NB + 255) / 256, 256, 0, stream>>>(rowsum, Lb);
}